// LocalQuantumAggregator_39238821216462
// MI455X (gfx1250) — compile-verified
//
#include <hip/hip_runtime.h>
#include <hip/hip_bf16.h>
#include <math.h>

typedef __attribute__((ext_vector_type(2))) float v2f;
typedef __attribute__((ext_vector_type(8))) float v8f;

__device__ __forceinline__ float2 cmul(float2 a, float2 b) {
  return make_float2(a.x * b.x - a.y * b.y, a.x * b.y + a.y * b.x);
}
__device__ __forceinline__ float2 cadd(float2 a, float2 b) {
  return make_float2(a.x + b.x, a.y + b.y);
}

// ---------------------------------------------------------------------------
// Kernel A: angles = all_f @ Wp.T + bp via V_WMMA_F32_16X16X4_F32.
// 256 rows (B*4) x 12 cols (padded 16), K=128 -> 32 WMMA steps per wave.
// Unconditional loads: B rows >= 12 are fetched from a clamped address and
// multiplied by a 0/1 mask (avoids exec-mask branches around each load).
// ---------------------------------------------------------------------------
__global__ void angles_wmma_kernel(const float* __restrict__ neigh,
                                   const float* __restrict__ selff,
                                   const float* __restrict__ Wp,
                                   const float* __restrict__ bp,
                                   float* __restrict__ angles /* 256 x 16 */) {
  const int lane  = threadIdx.x & 31;
  const int mrow  = lane & 15;
  const int khalf = lane >> 4;  // 0: K=0,1   1: K=2,3
  const int rowBase = blockIdx.x * 16;

  // A-fragment row: row = b*4 + s; s==0 -> self, else neighbor s-1
  const int r = rowBase + mrow;
  const int b = r >> 2, s = r & 3;
  const float* frow = (s == 0) ? (selff + b * 128)
                               : (neigh + (b * 3 + (s - 1)) * 128);
  // B[k][n] = Wp[n][k]; rows n>=12 are zero (clamped pointer + mask mult)
  const float wmask = (mrow < 12) ? 1.0f : 0.0f;
  const float* wrow = Wp + ((mrow < 12) ? mrow : 0) * 128;

  v8f acc = {};
#pragma unroll
  for (int k0 = 0; k0 < 128; k0 += 4) {
    v2f af, bf;
    af.x = frow[k0 + 2 * khalf + 0];
    af.y = frow[k0 + 2 * khalf + 1];
    bf.x = wrow[k0 + 2 * khalf + 0] * wmask;
    bf.y = wrow[k0 + 2 * khalf + 1] * wmask;
    acc = __builtin_amdgcn_wmma_f32_16x16x4_f32(
        false, af, false, bf, (short)0, acc, false, false);
  }

  const int ncol = lane & 15;
  const float bias = (ncol < 12) ? bp[ncol] : 0.0f;
#pragma unroll
  for (int j = 0; j < 8; ++j) {
    const int m = j + 8 * khalf;
    angles[(rowBase + m) * 16 + ncol] = acc[j] + bias;
  }
}

// ---------------------------------------------------------------------------
// Kernel B: fuse Rx*Ry*Rz (row-vector convention: s <- s*Rx*Ry*Rz).
// Block 0: per-(batch,qubit) fused layer-0 gate, row 0 only (state=|0>).
// Block 1: shared conv gates G0[16], G1[8] (full 2x2 complex).
// ---------------------------------------------------------------------------
__device__ __forceinline__ void rx_m(float th, float2 m[2][2]) {
  float c, sn; __sincosf(0.5f * th, &sn, &c);
  m[0][0] = make_float2(c, 0.f);   m[0][1] = make_float2(0.f, -sn);
  m[1][0] = make_float2(0.f, -sn); m[1][1] = make_float2(c, 0.f);
}
__device__ __forceinline__ void ry_m(float th, float2 m[2][2]) {
  float c, sn; __sincosf(0.5f * th, &sn, &c);
  m[0][0] = make_float2(c, 0.f);  m[0][1] = make_float2(-sn, 0.f);
  m[1][0] = make_float2(sn, 0.f); m[1][1] = make_float2(c, 0.f);
}
__device__ __forceinline__ void rz_m(float th, float2 m[2][2]) {
  float c, sn; __sincosf(0.5f * th, &sn, &c);
  m[0][0] = make_float2(c, -sn);   m[0][1] = make_float2(0.f, 0.f);
  m[1][0] = make_float2(0.f, 0.f); m[1][1] = make_float2(c, sn);
}
__device__ __forceinline__ void mm2(const float2 a[2][2], const float2 b[2][2],
                                    float2 o[2][2]) {
#pragma unroll
  for (int i = 0; i < 2; ++i)
#pragma unroll
    for (int j = 0; j < 2; ++j)
      o[i][j] = cadd(cmul(a[i][0], b[0][j]), cmul(a[i][1], b[1][j]));
}

__global__ void gateprep_kernel(const float* __restrict__ angles,
                                const float* __restrict__ rot0,
                                const float* __restrict__ rot1,
                                float* __restrict__ U0,
                                float* __restrict__ G0,
                                float* __restrict__ G1) {
  const int tid = threadIdx.x;
  float2 rx[2][2], ry[2][2], rz[2][2], t[2][2], m[2][2];
  if (blockIdx.x == 0) {
    const int b = tid >> 4, q = tid & 15;
    const float* ar = angles + (b * 4 + (q >> 2)) * 16 + (q & 3) * 3;
    rx_m(ar[0], rx); ry_m(ar[1], ry); rz_m(ar[2], rz);
    mm2(rx, ry, t); mm2(t, rz, m);
    float* o = U0 + (b * 16 + q) * 4;
    o[0] = m[0][0].x; o[1] = m[0][0].y; o[2] = m[0][1].x; o[3] = m[0][1].y;
  } else {
    const float* rot = (tid < 16) ? rot0 : rot1;
    float* G = (tid < 16) ? G0 : G1;
    const int q = (tid < 16) ? tid : tid - 16;
    if (tid < 24) {
      rx_m(rot[q * 3 + 0], rx); ry_m(rot[q * 3 + 1], ry); rz_m(rot[q * 3 + 2], rz);
      mm2(rx, ry, t); mm2(t, rz, m);
      float* o = G + q * 8;
      o[0] = m[0][0].x; o[1] = m[0][0].y; o[2] = m[0][1].x; o[3] = m[0][1].y;
      o[4] = m[1][0].x; o[5] = m[1][0].y; o[6] = m[1][1].x; o[7] = m[1][1].y;
    }
  }
}

// ---------------------------------------------------------------------------
// Kernel C1 (build): 256 blocks = (batch, quad) where quad = qubits 0..1.
// 256 threads/block (8 waves -> 2 waves/SIMD: ample VGPR budget for c[64]).
// x = quad<<14 | tid<<6 | r  (bit_q(x) = (x>>(15-q))&1)
//   qubits 10..15 -> r bits 5..0   : register-local 2x2
//   qubits  5..9  -> tid bits 4..0 : shfl_xor (lane)
//   qubits  2..4  -> tid bits 7..5 : LDS exchange (cross-wave)
//   qubits  0..1  -> quad          : handled in pool_kernel
// Builds |psi> as outer product of fused layer-0 gates, folds all 15 ZZ
// phases, applies conv0 gates on qubits 2..15, streams state to global.
// ---------------------------------------------------------------------------
#define LOADG(Gptr, q, g00, g01, g10, g11)                 \
  do {                                                     \
    const float* gp_ = (Gptr) + (q) * 8;                   \
    g00 = make_float2(gp_[0], gp_[1]);                     \
    g01 = make_float2(gp_[2], gp_[3]);                     \
    g10 = make_float2(gp_[4], gp_[5]);                     \
    g11 = make_float2(gp_[6], gp_[7]);                     \
  } while (0)

__global__ void __launch_bounds__(256)
build_kernel(const float* __restrict__ U0, const float* __restrict__ G0,
             const float* __restrict__ zz0, float2* __restrict__ state) {
  const int tid = threadIdx.x;        // 0..255
  const int lane = tid & 31;
  const int b = blockIdx.x >> 2;
  const int quad = blockIdx.x & 3;    // qubit0 = quad>>1, qubit1 = quad&1

  __shared__ float2 Htab[1024];       // prefix product, qubits 0..9
  __shared__ float2 Ping[1024];
  __shared__ float2 Ttab[64];         // prefix product, qubits 10..15
  __shared__ float2 Tp[64];
  __shared__ float ex_re[8 * 256];    // LDS exchange staging (8 slots)
  __shared__ float ex_im[8 * 256];

  // ---- H table (qubits 0..9) by doubling; 10 swaps -> ends in Htab ----
  if (tid == 0) Htab[0] = make_float2(1.f, 0.f);
  __syncthreads();
  {
    int sz = 1;
    float2* cur = Htab;
    float2* nxt = Ping;
    for (int q = 0; q < 10; ++q) {
      const float* up = U0 + (b * 16 + q) * 4;
      const float2 u0v = make_float2(up[0], up[1]);
      const float2 u1v = make_float2(up[2], up[3]);
      for (int i = tid; i < 2 * sz; i += 256)
        nxt[i] = cmul(cur[i >> 1], (i & 1) ? u1v : u0v);
      __syncthreads();
      float2* tp = cur; cur = nxt; nxt = tp;
      sz <<= 1;
    }
  }
  // fold ZZ pairs i=0..8 (both qubits in 0..9)
  for (int h = tid; h < 1024; h += 256) {
    float ph = 0.f;
#pragma unroll
    for (int i = 0; i < 9; ++i) {
      const int d = ((h >> (9 - i)) ^ (h >> (8 - i))) & 1;
      ph += (d ? 0.5f : -0.5f) * zz0[i];
    }
    float sp, cp; __sincosf(ph, &sp, &cp);
    Htab[h] = cmul(Htab[h], make_float2(cp, sp));
  }
  __syncthreads();

  // ---- T table (qubits 10..15); 6 swaps -> ends in Ttab ----
  if (tid == 0) Ttab[0] = make_float2(1.f, 0.f);
  __syncthreads();
  {
    int sz = 1;
    float2* cur = Ttab;
    float2* nxt = Tp;
    for (int q = 10; q < 16; ++q) {
      const float* up = U0 + (b * 16 + q) * 4;
      const float2 u0v = make_float2(up[0], up[1]);
      const float2 u1v = make_float2(up[2], up[3]);
      if (tid < 2 * sz) nxt[tid] = cmul(cur[tid >> 1], (tid & 1) ? u1v : u0v);
      __syncthreads();
      float2* tp = cur; cur = nxt; nxt = tp;
      sz <<= 1;
    }
  }
  if (tid < 64) {
    float ph = 0.f;
#pragma unroll
    for (int i = 10; i < 15; ++i) {
      const int d = ((tid >> (15 - i)) ^ (tid >> (14 - i))) & 1;
      ph += (d ? 0.5f : -0.5f) * zz0[i];
    }
    float sp, cp; __sincosf(ph, &sp, &cp);
    Ttab[tid] = cmul(Ttab[tid], make_float2(cp, sp));
  }
  __syncthreads();

  // ---- materialize: amp = H[quad:tid] * T[r] * boundary(zz0[9]) ----
  float c9, s9; __sincosf(0.5f * zz0[9], &s9, &c9);
  const float2 hv = Htab[(quad << 8) | tid];
  const int b9 = tid & 1;  // qubit 9 bit
  float2 c[64];
#pragma unroll
  for (int r = 0; r < 64; ++r) {
    float2 v = cmul(hv, Ttab[r]);
    const int df = b9 ^ ((r >> 5) & 1);  // qubit9 ^ qubit10
    c[r] = cmul(v, df ? make_float2(c9, s9) : make_float2(c9, -s9));
  }

  // ---- conv0 gates on qubits 10..15: register-local pairs ----
#pragma unroll
  for (int q = 10; q < 16; ++q) {
    const int m = 1 << (15 - q);
    float2 g00, g01, g10, g11;
    LOADG(G0, q, g00, g01, g10, g11);
#pragma unroll
    for (int r = 0; r < 64; ++r) {
      if (!(r & m)) {
        const float2 a = c[r], d2 = c[r | m];
        c[r]     = cadd(cmul(a, g00), cmul(d2, g10));
        c[r | m] = cadd(cmul(a, g01), cmul(d2, g11));
      }
    }
  }

  // ---- conv0 gates on qubits 5..9: cross-lane via shfl_xor ----
#pragma unroll
  for (int q = 5; q <= 9; ++q) {
    const int lm = 1 << (9 - q);
    float2 g00, g01, g10, g11;
    LOADG(G0, q, g00, g01, g10, g11);
    const int bit = (lane >> (9 - q)) & 1;
#pragma unroll
    for (int r = 0; r < 64; ++r) {
      const float2 own = c[r];
      float2 p;
      p.x = __shfl_xor(own.x, lm, 32);
      p.y = __shfl_xor(own.y, lm, 32);
      c[r] = bit ? cadd(cmul(p, g01), cmul(own, g11))
                 : cadd(cmul(own, g00), cmul(p, g10));
    }
  }

  // ---- conv0 gates on qubits 2..4: cross-wave via LDS exchange ----
#pragma unroll
  for (int q = 2; q <= 4; ++q) {
    const int tm = 1 << (9 - q);  // tid bits 7..5
    float2 g00, g01, g10, g11;
    LOADG(G0, q, g00, g01, g10, g11);
    const int bit = (tid >> (9 - q)) & 1;
    const int ptid = tid ^ tm;
#pragma unroll
    for (int base = 0; base < 64; base += 8) {
      __syncthreads();
#pragma unroll
      for (int j = 0; j < 8; ++j) {
        ex_re[j * 256 + tid] = c[base + j].x;
        ex_im[j * 256 + tid] = c[base + j].y;
      }
      __syncthreads();
#pragma unroll
      for (int j = 0; j < 8; ++j) {
        const float2 p = make_float2(ex_re[j * 256 + ptid], ex_im[j * 256 + ptid]);
        const float2 own = c[base + j];
        c[base + j] = bit ? cadd(cmul(p, g01), cmul(own, g11))
                          : cadd(cmul(own, g00), cmul(p, g10));
      }
    }
  }

  // ---- stream state to global (b128 stores, L2-resident) ----
  float4* outp = (float4*)(state + ((size_t)b << 16) + (quad << 14) + (tid << 6));
#pragma unroll
  for (int r = 0; r < 64; r += 2)
    outp[r >> 1] = make_float4(c[r].x, c[r].y, c[r + 1].x, c[r + 1].y);
}

// ---------------------------------------------------------------------------
// Kernel C2 (pool): 64 blocks (one per batch) x 1024 threads. Applies the
// conv0 gates on qubits 0..1 while streaming the state (L2-hot), does both
// poolings, conv1, measurement, and the output GEMM. Pass 2 recomputes the
// qubit-0/1 gates instead of caching state in registers.
// ---------------------------------------------------------------------------
__global__ void __launch_bounds__(1024)
pool_kernel(const float2* __restrict__ state, const float* __restrict__ G0,
            const float* __restrict__ G1, const float* __restrict__ zz1,
            const float* __restrict__ Wo, const float* __restrict__ bo,
            float* __restrict__ out) {
  const int tid = threadIdx.x;
  const int b = blockIdx.x;
  const float2* sb = state + ((size_t)b << 16);

  __shared__ float pP[256];
  __shared__ float sqp[256];
  __shared__ float2 Snew[256];
  __shared__ float2 S4[16];
  __shared__ float redv[2];
  __shared__ float zsh[4];

  float2 ga00, ga01, ga10, ga11;  // conv0 gate, qubit 0
  float2 gb00, gb01, gb10, gb11;  // conv0 gate, qubit 1
  LOADG(G0, 0, ga00, ga01, ga10, ga11);
  LOADG(G0, 1, gb00, gb01, gb10, gb11);

  if (tid < 256) { pP[tid] = 0.f; Snew[tid] = make_float2(0.f, 0.f); }
  __syncthreads();

  // ---- pass 1: |amp|^2 -> pP[t] ----
#pragma unroll
  for (int j = 0; j < 16; ++j) {
    const int g = j * 1024 + tid;  // x bits 13..0
    float2 a0 = sb[g];
    float2 a1 = sb[16384 + g];
    float2 a2 = sb[32768 + g];
    float2 a3 = sb[49152 + g];
    // qubit 0 (x bit 15): pairs (a0,a2), (a1,a3)
    float2 n0 = cadd(cmul(a0, ga00), cmul(a2, ga10));
    float2 n2 = cadd(cmul(a0, ga01), cmul(a2, ga11));
    float2 n1 = cadd(cmul(a1, ga00), cmul(a3, ga10));
    float2 n3 = cadd(cmul(a1, ga01), cmul(a3, ga11));
    // qubit 1 (x bit 14): pairs (n0,n1), (n2,n3)
    float2 m0 = cadd(cmul(n0, gb00), cmul(n1, gb10));
    float2 m1 = cadd(cmul(n0, gb01), cmul(n1, gb11));
    float2 m2 = cadd(cmul(n2, gb00), cmul(n3, gb10));
    float2 m3 = cadd(cmul(n2, gb01), cmul(n3, gb11));
    const int tlow = (((g >> 12) & 1) << 6) | (((g >> 10) & 1) << 5) |
                     (((g >> 8) & 1) << 4) | (((g >> 6) & 1) << 3) |
                     (((g >> 4) & 1) << 2) | (((g >> 2) & 1) << 1) | (g & 1);
    atomicAdd(&pP[tlow],        m0.x * m0.x + m0.y * m0.y);   // qq=0: t7=0
    atomicAdd(&pP[tlow | 128],  m1.x * m1.x + m1.y * m1.y);   // qq=1: t7=1
    atomicAdd(&pP[tlow],        m2.x * m2.x + m2.y * m2.y);   // qq=2: t7=0
    atomicAdd(&pP[tlow | 128],  m3.x * m3.x + m3.y * m3.y);   // qq=3: t7=1
  }
  __syncthreads();
  if (tid == 0) {
    float tot = 0.f;
    for (int i = 0; i < 256; ++i) tot += pP[i];
    redv[0] = tot;
  }
  __syncthreads();
  if (tid < 256) sqp[tid] = sqrtf(pP[tid] / (redv[0] + 1e-10f));
  __syncthreads();

  // ---- pass 2: recompute gates, Snew[k] += amp * sqrt(p[t]) ----
#pragma unroll
  for (int j = 0; j < 16; ++j) {
    const int g = j * 1024 + tid;
    float2 a0 = sb[g];
    float2 a1 = sb[16384 + g];
    float2 a2 = sb[32768 + g];
    float2 a3 = sb[49152 + g];
    float2 n0 = cadd(cmul(a0, ga00), cmul(a2, ga10));
    float2 n2 = cadd(cmul(a0, ga01), cmul(a2, ga11));
    float2 n1 = cadd(cmul(a1, ga00), cmul(a3, ga10));
    float2 n3 = cadd(cmul(a1, ga01), cmul(a3, ga11));
    float2 m0 = cadd(cmul(n0, gb00), cmul(n1, gb10));
    float2 m1 = cadd(cmul(n0, gb01), cmul(n1, gb11));
    float2 m2 = cadd(cmul(n2, gb00), cmul(n3, gb10));
    float2 m3 = cadd(cmul(n2, gb01), cmul(n3, gb11));
    const int tlow = (((g >> 12) & 1) << 6) | (((g >> 10) & 1) << 5) |
                     (((g >> 8) & 1) << 4) | (((g >> 6) & 1) << 3) |
                     (((g >> 4) & 1) << 2) | (((g >> 2) & 1) << 1) | (g & 1);
    const int klow = (((g >> 13) & 1) << 6) | (((g >> 11) & 1) << 5) |
                     (((g >> 9) & 1) << 4) | (((g >> 7) & 1) << 3) |
                     (((g >> 5) & 1) << 2) | (((g >> 3) & 1) << 1) |
                     ((g >> 1) & 1);
    const float w0 = sqp[tlow], w1 = sqp[tlow | 128];
    // qq bit1 = qubit0 -> k bit 7 ; qq bit0 = qubit1 -> t bit 7
    atomicAdd(&Snew[klow].x,       m0.x * w0);
    atomicAdd(&Snew[klow].y,       m0.y * w0);
    atomicAdd(&Snew[klow].x,       m1.x * w1);
    atomicAdd(&Snew[klow].y,       m1.y * w1);
    atomicAdd(&Snew[klow | 128].x, m2.x * w0);
    atomicAdd(&Snew[klow | 128].y, m2.y * w0);
    atomicAdd(&Snew[klow | 128].x, m3.x * w1);
    atomicAdd(&Snew[klow | 128].y, m3.y * w1);
  }
  __syncthreads();
  if (tid == 0) {
    float n2 = 0.f;
    for (int i = 0; i < 256; ++i)
      n2 += Snew[i].x * Snew[i].x + Snew[i].y * Snew[i].y;
    redv[1] = rsqrtf(n2 + 1e-10f);
  }
  __syncthreads();
  if (tid < 256) {
    Snew[tid].x *= redv[1];
    Snew[tid].y *= redv[1];
  }
  __syncthreads();

  // ---- conv1 on 8 qubits (state in LDS) ----
  if (tid < 256) {
    float ph = 0.f;
#pragma unroll
    for (int i = 0; i < 7; ++i) {
      const int d = ((tid >> (7 - i)) ^ (tid >> (6 - i))) & 1;
      ph += (d ? 0.5f : -0.5f) * zz1[i];
    }
    float sp, cp; __sincosf(ph, &sp, &cp);
    Snew[tid] = cmul(Snew[tid], make_float2(cp, sp));
  }
  __syncthreads();
#pragma unroll
  for (int q = 0; q < 8; ++q) {
    float2 g00, g01, g10, g11;
    LOADG(G1, q, g00, g01, g10, g11);
    const int m = 1 << (7 - q);
    if (tid < 128) {
      const int i0 = ((tid & ~(m - 1)) << 1) | (tid & (m - 1));
      const int i1 = i0 | m;
      const float2 a = Snew[i0], d2 = Snew[i1];
      Snew[i0] = cadd(cmul(a, g00), cmul(d2, g10));
      Snew[i1] = cadd(cmul(a, g01), cmul(d2, g11));
    }
    __syncthreads();
  }

  // ---- pooling 8 -> 4 qubits ----
  if (tid < 16) { pP[tid] = 0.f; S4[tid] = make_float2(0.f, 0.f); }
  __syncthreads();
  if (tid < 256) {
    const int tt = (((tid >> 6) & 1) << 3) | (((tid >> 4) & 1) << 2) |
                   (((tid >> 2) & 1) << 1) | (tid & 1);
    atomicAdd(&pP[tt], Snew[tid].x * Snew[tid].x + Snew[tid].y * Snew[tid].y);
  }
  __syncthreads();
  if (tid == 0) {
    float tot = 0.f;
    for (int i = 0; i < 16; ++i) tot += pP[i];
    redv[0] = tot;
  }
  __syncthreads();
  if (tid < 16) sqp[tid] = sqrtf(pP[tid] / (redv[0] + 1e-10f));
  __syncthreads();
  if (tid < 256) {
    const int kk = (((tid >> 7) & 1) << 3) | (((tid >> 5) & 1) << 2) |
                   (((tid >> 3) & 1) << 1) | ((tid >> 1) & 1);
    const int tt = (((tid >> 6) & 1) << 3) | (((tid >> 4) & 1) << 2) |
                   (((tid >> 2) & 1) << 1) | (tid & 1);
    const float w = sqp[tt];
    atomicAdd(&S4[kk].x, Snew[tid].x * w);
    atomicAdd(&S4[kk].y, Snew[tid].y * w);
  }
  __syncthreads();
  if (tid == 0) {
    float n2 = 0.f;
    for (int i = 0; i < 16; ++i) n2 += S4[i].x * S4[i].x + S4[i].y * S4[i].y;
    redv[1] = rsqrtf(n2 + 1e-10f);
  }
  __syncthreads();
  if (tid < 16) {
    S4[tid].x *= redv[1];
    S4[tid].y *= redv[1];
  }
  __syncthreads();

  // ---- measure Z on 4 qubits ----
  if (tid < 4) {
    float z = 0.f;
#pragma unroll
    for (int x = 0; x < 16; ++x) {
      const float m2 = S4[x].x * S4[x].x + S4[x].y * S4[x].y;
      z += ((x >> (3 - tid)) & 1) ? -m2 : m2;
    }
    zsh[tid] = z;
  }
  __syncthreads();

  // ---- out[b] = z @ Wo.T + bo ----
  if (tid < 64) {
    float acc = bo[tid];
#pragma unroll
    for (int j = 0; j < 4; ++j) acc += zsh[j] * Wo[tid * 4 + j];
    out[b * 64 + tid] = acc;
  }
}

// ---------------------------------------------------------------------------
extern "C" void kernel_launch(void* const* d_in, const int* in_sizes, int n_in,
                              void* d_out, int out_size, void* d_ws, size_t ws_size,
                              hipStream_t stream) {
  const float* neigh = (const float*)d_in[0];  // (64,3,128)
  const float* selff = (const float*)d_in[1];  // (64,128)
  const float* Wp    = (const float*)d_in[2];  // (12,128)
  const float* bp    = (const float*)d_in[3];  // (12,)
  const float* rot0  = (const float*)d_in[4];  // (16,3)
  const float* zz0   = (const float*)d_in[5];  // (15,)
  const float* rot1  = (const float*)d_in[6];  // (8,3)
  const float* zz1   = (const float*)d_in[7];  // (7,)
  const float* Wo    = (const float*)d_in[8];  // (64,4)
  const float* bo    = (const float*)d_in[9];  // (64,)

  float* ws     = (float*)d_ws;
  float* angles = ws;            // 256*16 floats
  float* U0     = ws + 4096;     // 64*16*4
  float* G0     = ws + 8192;     // 16*8
  float* G1     = ws + 8320;     // 8*8
  float2* state = (float2*)(ws + 16384);  // 64 * 65536 complexes = 32 MB

  angles_wmma_kernel<<<16, 32, 0, stream>>>(neigh, selff, Wp, bp, angles);
  gateprep_kernel<<<2, 1024, 0, stream>>>(angles, rot0, rot1, U0, G0, G1);
  build_kernel<<<256, 256, 0, stream>>>(U0, G0, zz0, state);
  pool_kernel<<<64, 1024, 0, stream>>>(state, G0, G1, zz1, Wo, bo,
                                       (float*)d_out);
}